// TanhAttention_66537633350264
// MI455X (gfx1250) — compile-verified
//
#include <hip/hip_runtime.h>
#include <hip/hip_bf16.h>

typedef __attribute__((ext_vector_type(2))) float v2f;
typedef __attribute__((ext_vector_type(8))) float v8f;

#define BATCH 4
#define L 256   // Lx == Lm == d == 256
#define DD 256
#define SSTR 260  // padded LDS row stride (floats): 260 % 64 == 4 -> conflict-free

__device__ __forceinline__ float fast_tanh(float v) {
#if __has_builtin(__builtin_amdgcn_tanhf)
    return __builtin_amdgcn_tanhf(v);     // v_tanh_f32 on gfx1250
#else
    return tanhf(v);
#endif
}

__device__ __forceinline__ v8f wmma_f32_k4(v2f a, v2f b, v8f c) {
#if __has_builtin(__builtin_amdgcn_wmma_f32_16x16x4_f32)
    return __builtin_amdgcn_wmma_f32_16x16x4_f32(false, a, false, b, (short)0, c,
                                                 false, false);
#else
    // placeholder fallback (keeps compile alive; signals wmma=0 in histogram)
    c[0] += a[0] * b[0] + a[1] * b[1];
    return c;
#endif
}

// ---------------------------------------------------------------------------
// Kernel A: item1 = x @ W1^T + b1 ; item2 = memory @ W2^T  (NT GEMMs, fp32 WMMA)
// 2048 waves, one 16x16 output tile each; K=256 in steps of 4.
// ---------------------------------------------------------------------------
__global__ void proj_kernel(const float* __restrict__ x, const float* __restrict__ mem,
                            const float* __restrict__ w1, const float* __restrict__ b1,
                            const float* __restrict__ w2,
                            float* __restrict__ item1, float* __restrict__ item2) {
    int wave = (blockIdx.x * blockDim.x + threadIdx.x) >> 5;   // 0..2047
    int lane = threadIdx.x & 31;
    int half = lane >> 4, l16 = lane & 15;

    bool is1 = wave < 1024;                 // wave-uniform
    int t  = is1 ? wave : wave - 1024;      // tile within matrix
    int rt = t >> 4;                        // 64 row tiles of [1024 x 256]
    int ct = t & 15;                        // 16 col tiles

    const float* A = is1 ? x  : mem;        // [1024, 256], K contiguous
    const float* W = is1 ? w1 : w2;         // [256, 256],  K contiguous (NT)
    float* Dst     = is1 ? item1 : item2;

    const float* Aptr = A + (rt * 16 + l16) * DD + 2 * half;   // M = l16
    const float* Wptr = W + (ct * 16 + l16) * DD + 2 * half;   // N = l16

    v8f acc = {};
#pragma unroll 8
    for (int k0 = 0; k0 < DD; k0 += 4) {
        v2f a = *(const v2f*)(Aptr + k0);   // K = k0 + 2*half + {0,1}
        v2f b = *(const v2f*)(Wptr + k0);
        acc = wmma_f32_k4(a, b, acc);
    }

    float bias = is1 ? b1[ct * 16 + l16] : 0.0f;   // bias indexed by N
#pragma unroll
    for (int r = 0; r < 8; ++r) {
        int m = r + 8 * half;               // C/D row layout
        Dst[(rt * 16 + m) * DD + ct * 16 + l16] = acc[r] + bias;
    }
}

// ---------------------------------------------------------------------------
// Kernel B: S[b,x,m] = sum_d wst[d] * tanh(item1[b,x,d] + item2[b,m,d])
// One block per 16x16 (x,m) tile; tiles staged in padded LDS; v_tanh_f32 inner loop.
// ---------------------------------------------------------------------------
__global__ void score_kernel(const float* __restrict__ item1, const float* __restrict__ item2,
                             const float* __restrict__ wst, float* __restrict__ S) {
    __shared__ float s1[16 * SSTR];
    __shared__ float s2[16 * SSTR];
    __shared__ float sw[DD];

    int b = blockIdx.z;
    int xt = blockIdx.x * 16, mt = blockIdx.y * 16;
    int tid = threadIdx.x;                  // 256 threads

    int row = tid >> 4;                     // 0..15
    int seg = (tid & 15) * 16;              // 16-float segment
    const float4* g1 = (const float4*)(item1 + ((b * L) + xt + row) * DD + seg);
    const float4* g2 = (const float4*)(item2 + ((b * L) + mt + row) * DD + seg);
    float4* l1 = (float4*)(s1 + row * SSTR + seg);
    float4* l2 = (float4*)(s2 + row * SSTR + seg);
#pragma unroll
    for (int j = 0; j < 4; ++j) { l1[j] = g1[j]; l2[j] = g2[j]; }
    sw[tid] = wst[tid];
    __syncthreads();

    int tm = tid & 15;                      // contiguous store dim
    int tx = tid >> 4;
    const float4* R1 = (const float4*)(s1 + tx * SSTR);
    const float4* R2 = (const float4*)(s2 + tm * SSTR);
    const float4* RW = (const float4*)sw;

    float acc = 0.0f;
#pragma unroll 4
    for (int d4 = 0; d4 < DD / 4; ++d4) {
        float4 a = R1[d4], c = R2[d4], w = RW[d4];
        acc += w.x * fast_tanh(a.x + c.x);
        acc += w.y * fast_tanh(a.y + c.y);
        acc += w.z * fast_tanh(a.z + c.z);
        acc += w.w * fast_tanh(a.w + c.w);
    }
    S[((b * L) + xt + tx) * L + mt + tm] = acc;
}

// ---------------------------------------------------------------------------
// Kernel C: softmaxes. z=0: P = softmax_m(mask ? S : -1e30), also emits masked S.
//           z=1: Q[b,m,x] = softmax_x(S[b, :, m])  (UNMASKED, per reference).
// ---------------------------------------------------------------------------
__global__ void softmax_kernel(const float* __restrict__ S, const int* __restrict__ mask,
                               float* __restrict__ P, float* __restrict__ Q,
                               float* __restrict__ Smasked) {
    __shared__ float red[256];
    int b = blockIdx.y, row = blockIdx.x, mode = blockIdx.z;
    int t = threadIdx.x;

    float val;
    if (mode == 0) {                        // row = x, t = m (masked)
        float s = S[((b * L) + row) * L + t];
        val = (mask[b * L + t] != 0) ? s : -1e30f;
        Smasked[((b * L) + row) * L + t] = val;
    } else {                                // row = m, t = x (unmasked, transposed read)
        val = S[((b * L) + t) * L + row];
    }

    red[t] = val; __syncthreads();
#pragma unroll
    for (int off = 128; off >= 1; off >>= 1) {
        if (t < off) red[t] = fmaxf(red[t], red[t + off]);
        __syncthreads();
    }
    float mx = red[0]; __syncthreads();

    float e = __expf(val - mx);
    red[t] = e; __syncthreads();
#pragma unroll
    for (int off = 128; off >= 1; off >>= 1) {
        if (t < off) red[t] += red[t + off];
        __syncthreads();
    }
    float p = e / red[0];

    if (mode == 0) P[((b * L) + row) * L + t] = p;
    else           Q[((b * L) + row) * L + t] = p;
}

// ---------------------------------------------------------------------------
// Kernel D: out = P @ memory ; agg_2_h = Q @ x   (batched NN GEMMs, fp32 WMMA)
// ---------------------------------------------------------------------------
__global__ void readout_kernel(const float* __restrict__ P, const float* __restrict__ Q,
                               const float* __restrict__ mem, const float* __restrict__ x,
                               float* __restrict__ out, float* __restrict__ agg) {
    int wave = (blockIdx.x * blockDim.x + threadIdx.x) >> 5;   // 0..2047
    int lane = threadIdx.x & 31;
    int half = lane >> 4, l16 = lane & 15;

    bool first = wave < 1024;               // wave-uniform
    int t = first ? wave : wave - 1024;
    int bidx = t >> 8;                      // batch
    int tt = t & 255;
    int rt = tt >> 4, ct = tt & 15;

    const float* A  = (first ? P   : Q) + bidx * L * L;   // [256,256] K contiguous
    const float* Bm = (first ? mem : x) + bidx * L * DD;  // [K=256, N=256] N contiguous
    float* Dp       = (first ? out : agg) + bidx * L * DD;

    const float* Aptr = A + (rt * 16 + l16) * L + 2 * half;           // M = l16
    const float* Bptr = Bm + (2 * half) * DD + ct * 16 + l16;         // N = l16

    v8f acc = {};
#pragma unroll 8
    for (int k0 = 0; k0 < L; k0 += 4) {
        v2f a = *(const v2f*)(Aptr + k0);
        v2f b;
        b[0] = Bptr[(k0 + 0) * DD];         // K = k0 + 2*half + 0
        b[1] = Bptr[(k0 + 1) * DD];         // K = k0 + 2*half + 1
        acc = wmma_f32_k4(a, b, acc);
    }
#pragma unroll
    for (int r = 0; r < 8; ++r) {
        int m = r + 8 * half;
        Dp[(rt * 16 + m) * DD + ct * 16 + l16] = acc[r];
    }
}

extern "C" void kernel_launch(void* const* d_in, const int* in_sizes, int n_in,
                              void* d_out, int out_size, void* d_ws, size_t ws_size,
                              hipStream_t stream) {
    const float* x      = (const float*)d_in[0];   // [4,256,256]
    const float* memory = (const float*)d_in[1];   // [4,256,256]
    const int*   mask   = (const int*)  d_in[2];   // [4,256]
    const float* w1     = (const float*)d_in[3];   // [256,256]
    const float* b1     = (const float*)d_in[4];   // [256]
    const float* w2     = (const float*)d_in[5];   // [256,256]
    const float* wst    = (const float*)d_in[6];   // [1,256]

    float* out_p = (float*)d_out;                  // out:  [4,256,256]
    float* S_p   = out_p + BATCH * L * L;          // S:    [4,256,256] (masked)
    float* agg_p = S_p + BATCH * L * L;            // agg:  [4,256,256]

    float* ws    = (float*)d_ws;                   // needs 3 MB
    float* item1 = ws;                             // [4*256,256]
    float* item2 = ws + BATCH * L * DD;            // [4*256,256]
    float* Sws   = ws + 2 * BATCH * L * DD;        // [4,256,256] unmasked S
    float* Pw    = item1;                          // reuse after score pass
    float* Qw    = item2;

    proj_kernel   <<<256, 256, 0, stream>>>(x, memory, w1, b1, w2, item1, item2);
    score_kernel  <<<dim3(16, 16, BATCH), 256, 0, stream>>>(item1, item2, wst, Sws);
    softmax_kernel<<<dim3(L, BATCH, 2), 256, 0, stream>>>(Sws, mask, Pw, Qw, S_p);
    readout_kernel<<<256, 256, 0, stream>>>(Pw, Qw, memory, x, out_p, agg_p);
}